// EnhancedSocialRecommender_87866440942265
// MI455X (gfx1250) — compile-verified
//
#include <hip/hip_runtime.h>
#include <hip/hip_bf16.h>
#include <math.h>

// ---------------------------------------------------------------------------
// EnhancedSocialRecommender for MI455X (gfx1250, wave32)
//  - SpMM phases: edge-parallel waves, global_atomic_add_f32 (tables L2-resident)
//  - Dense [N,64]@[K,64] GEMMs: V_WMMA_F32_16X16X4_F32, one 16x16 tile/wave,
//    W staged to LDS via async global->LDS copies (ASYNCcnt path)
//  - LN / leaky / sigmoid-gate / scoring: one wave32 per row, shfl_xor reductions
// ---------------------------------------------------------------------------

typedef __attribute__((ext_vector_type(2))) float v2f;
typedef __attribute__((ext_vector_type(8))) float v8f;

#define D 64
#define LRELU 0.01f
#define LN_EPS 1e-5f

__device__ __forceinline__ float waveReduceSum(float v) {
#pragma unroll
  for (int off = 16; off > 0; off >>= 1) v += __shfl_xor(v, off, 32);
  return v;
}

// -------------------- tiny softmax of layer_weights[0..2] -------------------
__global__ void esr_compute_nw(const float* __restrict__ lw, float* __restrict__ nw) {
  if (threadIdx.x == 0 && blockIdx.x == 0) {
    float m = fmaxf(lw[0], fmaxf(lw[1], lw[2]));
    float e0 = expf(lw[0] - m), e1 = expf(lw[1] - m), e2 = expf(lw[2] - m);
    float inv = 1.0f / (e0 + e1 + e2);
    nw[0] = e0 * inv; nw[1] = e1 * inv; nw[2] = e2 * inv;
  }
}

// -------------------- elementwise helpers -----------------------------------
__global__ void esr_scale_init(const float* __restrict__ in, float* __restrict__ out,
                               float s, size_t n) {
  size_t i = (size_t)blockIdx.x * blockDim.x + threadIdx.x;
  if (i < n) out[i] = s * in[i];
}

__global__ void esr_copy(const float* __restrict__ in, float* __restrict__ out, size_t n) {
  size_t i = (size_t)blockIdx.x * blockDim.x + threadIdx.x;
  if (i < n) out[i] = in[i];
}

// out = nw0*a + nw1*b + nw2*c
__global__ void esr_weighted3(const float* __restrict__ a, const float* __restrict__ b,
                              const float* __restrict__ c, const float* __restrict__ nw,
                              float* __restrict__ out, size_t n) {
  size_t i = (size_t)blockIdx.x * blockDim.x + threadIdx.x;
  if (i < n) out[i] = nw[0] * a[i] + nw[1] * b[i] + nw[2] * c[i];
}

// out = nw[idx]*in
__global__ void esr_scale_by_nw(const float* __restrict__ in, const float* __restrict__ nw,
                                int idx, float* __restrict__ out, size_t n) {
  size_t i = (size_t)blockIdx.x * blockDim.x + threadIdx.x;
  if (i < n) out[i] = nw[idx] * in[i];
}

// -------------------- SpMM: out[dst[e]] += w[e]*x[src[e]] -------------------
// one wave per edge; each lane covers dims {lane, lane+32}
__global__ void esr_spmm(const int* __restrict__ dst, const int* __restrict__ src,
                         const float* __restrict__ w, const float* __restrict__ x,
                         float* __restrict__ out, int nnz) {
  int t = blockIdx.x * blockDim.x + threadIdx.x;
  int e = t >> 5;
  int lane = t & 31;
  if (e >= nnz) return;
  float wv = w[e];
  const float* xr = x + (size_t)src[e] * D;
  float* orow = out + (size_t)dst[e] * D;
  atomicAdd(orow + lane,      wv * xr[lane]);
  atomicAdd(orow + lane + 32, wv * xr[lane + 32]);
}

// -------------------- WMMA GEMM: C[N,64] = concat(A1,A2)[N,K] @ W[K,64] ----
// K = 64 (A2 unused) or 128 (cols 64..127 read from A2). nRows % 32 == 0.
// Block = 256 threads = 8 waves: 2 row-tiles x 4 col-tiles of 16x16 outputs.
// W is staged into LDS once per block with async global->LDS b128 copies.
__global__ void esr_gemm_wmma(const float* __restrict__ A1, const float* __restrict__ A2,
                              const float* __restrict__ W, float* __restrict__ C,
                              int nRows, int K) {
  __shared__ float Wlds[128 * D];           // up to 32 KB (WGP has 320 KB)

  // ---- stage W[K,64] into LDS (async path, tracked by ASYNCcnt)
  const int kw = K * D;                      // 4096 or 8192 floats
  for (int off = threadIdx.x * 4; off < kw; off += 256 * 4) {
    unsigned lds_addr = (unsigned)(size_t)(&Wlds[off]);
    const float* gaddr = W + off;
    asm volatile("global_load_async_to_lds_b128 %0, %1, off"
                 :: "v"(lds_addr), "v"(gaddr) : "memory");
  }
  asm volatile("s_wait_asynccnt 0x0" ::: "memory");
  __syncthreads();

  int wave = threadIdx.x >> 5;              // 0..7
  int lane = threadIdx.x & 31;
  int rowTile = blockIdx.x * 2 + (wave >> 2);
  int colTile = wave & 3;                   // 4 tiles of 16 cols
  int rowBase = rowTile * 16;
  if (rowBase < nRows) {
    int hi = lane >> 4;                     // 0|1  (K sub-pair select)
    int lo = lane & 15;                     // M (for A) / N (for B,C,D)
    size_t m = (size_t)(rowBase + lo);
    const float* Wl = &Wlds[colTile * 16 + lo];

    v8f c = {};
#pragma unroll
    for (int k0 = 0; k0 < 64; k0 += 4) {    // first 64 cols from A1
      int ka = k0 + 2 * hi;
      v2f a; a.x = A1[m * D + ka]; a.y = A1[m * D + ka + 1];
      v2f b; b.x = Wl[(size_t)ka * D]; b.y = Wl[(size_t)(ka + 1) * D];
      c = __builtin_amdgcn_wmma_f32_16x16x4_f32(
          false, a, false, b, (short)0, c, false, false);
    }
    if (K > 64) {
#pragma unroll
      for (int k0 = 64; k0 < 128; k0 += 4) { // cols 64..127 from A2
        int ka = k0 + 2 * hi;
        v2f a; a.x = A2[m * D + (ka - 64)]; a.y = A2[m * D + (ka - 63)];
        v2f b; b.x = Wl[(size_t)ka * D]; b.y = Wl[(size_t)(ka + 1) * D];
        c = __builtin_amdgcn_wmma_f32_16x16x4_f32(
            false, a, false, b, (short)0, c, false, false);
      }
    }
    // C/D layout: VGPR r holds row M = r + 8*hi, col N = lo
    float* Crow = C + (size_t)(rowBase + 8 * hi) * D + colTile * 16 + lo;
#pragma unroll
    for (int r = 0; r < 8; ++r) Crow[(size_t)r * D] = c[r];
  }
}

// -------------------- social layer epilogue ---------------------------------
// gate = sigmoid(leaky(LN(gpre + b, g, beta)));  curOut = gate*new + (1-gate)*curIn
// also socialW += nw[nwIdx] * curOut.  one wave per row.
__global__ void esr_social_epilogue(const float* __restrict__ gpre,
                                    const float* __restrict__ bias,
                                    const float* __restrict__ g,
                                    const float* __restrict__ beta,
                                    const float* __restrict__ newbuf,
                                    const float* __restrict__ curIn,
                                    float* __restrict__ curOut,
                                    float* __restrict__ socialW,
                                    const float* __restrict__ nw, int nwIdx,
                                    int nRows) {
  int t = blockIdx.x * blockDim.x + threadIdx.x;
  int row = t >> 5;
  int lane = t & 31;
  if (row >= nRows) return;
  size_t base = (size_t)row * D;
  int d0 = lane, d1 = lane + 32;
  float x0 = gpre[base + d0] + bias[d0];
  float x1 = gpre[base + d1] + bias[d1];
  float mu = waveReduceSum(x0 + x1) * (1.0f / D);
  float v0 = x0 - mu, v1 = x1 - mu;
  float var = waveReduceSum(v0 * v0 + v1 * v1) * (1.0f / D);
  float rs = rsqrtf(var + LN_EPS);
  float wnw = nw[nwIdx];
#pragma unroll
  for (int k = 0; k < 2; ++k) {
    int dd = (k == 0) ? d0 : d1;
    float vv = (k == 0) ? v0 : v1;
    float y = vv * rs * g[dd] + beta[dd];
    y = (y >= 0.0f) ? y : LRELU * y;          // leaky_relu
    float gate = 1.0f / (1.0f + expf(-y));    // sigmoid
    float nv = newbuf[base + dd];
    float cv = curIn[base + dd];
    float o = gate * nv + (1.0f - gate) * cv;
    curOut[base + dd] = o;
    socialW[base + dd] += wnw * o;
  }
}

// -------------------- metapath epilogue: leaky(LN(raw + b, g, beta)) --------
__global__ void esr_mp_epilogue(const float* __restrict__ raw,
                                const float* __restrict__ bias,
                                const float* __restrict__ g,
                                const float* __restrict__ beta,
                                float* __restrict__ out, int nRows) {
  int t = blockIdx.x * blockDim.x + threadIdx.x;
  int row = t >> 5;
  int lane = t & 31;
  if (row >= nRows) return;
  size_t base = (size_t)row * D;
  int d0 = lane, d1 = lane + 32;
  float x0 = raw[base + d0] + bias[d0];
  float x1 = raw[base + d1] + bias[d1];
  float mu = waveReduceSum(x0 + x1) * (1.0f / D);
  float v0 = x0 - mu, v1 = x1 - mu;
  float var = waveReduceSum(v0 * v0 + v1 * v1) * (1.0f / D);
  float rs = rsqrtf(var + LN_EPS);
  float y0 = v0 * rs * g[d0] + beta[d0];
  float y1 = v1 * rs * g[d1] + beta[d1];
  out[base + d0] = (y0 >= 0.0f) ? y0 : LRELU * y0;
  out[base + d1] = (y1 >= 0.0f) ? y1 : LRELU * y1;
}

// -------------------- scoring: dot(fused[u], item_weighted[p]) --------------
__global__ void esr_score(const int* __restrict__ users, const int* __restrict__ pos,
                          const float* __restrict__ fused,
                          const float* __restrict__ i0, const float* __restrict__ i1,
                          const float* __restrict__ i2, const float* __restrict__ nw,
                          float* __restrict__ scores, int B) {
  int t = blockIdx.x * blockDim.x + threadIdx.x;
  int b = t >> 5;
  int lane = t & 31;
  if (b >= B) return;
  const float* fu = fused + (size_t)users[b] * D;
  size_t p = (size_t)pos[b] * D;
  float n0 = nw[0], n1 = nw[1], n2 = nw[2];
  float acc = 0.0f;
#pragma unroll
  for (int k = 0; k < 2; ++k) {
    int dd = lane + 32 * k;
    float iw = n0 * i0[p + dd] + n1 * i1[p + dd] + n2 * i2[p + dd];
    acc += fu[dd] * iw;
  }
  acc = waveReduceSum(acc);
  if (lane == 0) scores[b] = acc;
}

// ---------------------------------------------------------------------------
extern "C" void kernel_launch(void* const* d_in, const int* in_sizes, int n_in,
                              void* d_out, int out_size, void* d_ws, size_t ws_size,
                              hipStream_t stream) {
  const int* users     = (const int*)d_in[0];
  const int* pos_items = (const int*)d_in[1];
  const int* ui_src    = (const int*)d_in[2];
  const int* ui_dst    = (const int*)d_in[3];
  const float* ui_w    = (const float*)d_in[4];
  const int* s_src     = (const int*)d_in[5];
  const int* s_dst     = (const int*)d_in[6];
  const float* s_w     = (const float*)d_in[7];
  const float* u0      = (const float*)d_in[8];
  const float* i0      = (const float*)d_in[9];
  const float* lw      = (const float*)d_in[10];
  const float* Wsoc    = (const float*)d_in[11];
  const float* ho_W    = (const float*)d_in[12];
  const float* ho_b    = (const float*)d_in[13];
  const float* ho_g    = (const float*)d_in[14];
  const float* ho_beta = (const float*)d_in[15];
  const float* mp_W    = (const float*)d_in[16];
  const float* mp_b    = (const float*)d_in[17];
  const float* mp_g    = (const float*)d_in[18];
  const float* mp_beta = (const float*)d_in[19];

  const int B      = in_sizes[0];
  const int NNZ_UI = in_sizes[2];
  const int NNZ_S  = in_sizes[5];
  const int U      = in_sizes[8] / D;   // 100000 (divisible by 32)
  const int I      = in_sizes[9] / D;   // 150000
  const size_t UD = (size_t)U * D;
  const size_t ID = (size_t)I * D;

  // workspace carve-up (floats)
  float* ws   = (float*)d_ws;
  float* nw   = ws;                 // 16 floats
  float* u1   = ws + 16;            // [U,D]
  float* u2   = u1 + UD;            // [U,D]  (= user_last = s_list[0])
  float* uw   = u2 + UD;            // user_weighted
  float* sw   = uw + UD;            // social_weighted accumulator
  float* cur1 = sw + UD;            // social layer-1 output
  float* cur2 = cur1 + UD;          // social layer-2 output
  float* lin  = cur2 + UD;          // Wsoc output / gpre / fusedRaw (reused)
  float* nbuf = lin + UD;           // "new" buffer / fused output (reused)
  float* i1   = nbuf + UD;          // [I,D]
  float* i2   = i1 + ID;            // [I,D]
  (void)ws_size; (void)n_in;

  const int TB = 256;
  const int gU  = (int)((UD + TB - 1) / TB);
  const int gI  = (int)((ID + TB - 1) / TB);
  const int gEu = (NNZ_UI * 32 + TB - 1) / TB;   // wave-per-edge
  const int gEs = (NNZ_S  * 32 + TB - 1) / TB;
  const int gRow = (U * 32 + TB - 1) / TB;       // wave-per-row epilogues
  const int gGemm = (U / 16 + 1) / 2;            // 2 row-tiles per block

  // softmax of layer weights
  esr_compute_nw<<<1, 32, 0, stream>>>(lw, nw);

  // ---- LightGCN layer 1
  esr_scale_init<<<gU, TB, 0, stream>>>(u0, u1, 0.1f, UD);
  esr_scale_init<<<gI, TB, 0, stream>>>(i0, i1, 0.1f, ID);
  esr_spmm<<<gEu, TB, 0, stream>>>(ui_src, ui_dst, ui_w, i0, u1, NNZ_UI);
  esr_spmm<<<gEu, TB, 0, stream>>>(ui_dst, ui_src, ui_w, u0, i1, NNZ_UI);

  // ---- LightGCN layer 2
  esr_scale_init<<<gU, TB, 0, stream>>>(u0, u2, 0.1f, UD);
  esr_scale_init<<<gI, TB, 0, stream>>>(i0, i2, 0.1f, ID);
  esr_spmm<<<gEu, TB, 0, stream>>>(ui_src, ui_dst, ui_w, i1, u2, NNZ_UI);
  esr_spmm<<<gEu, TB, 0, stream>>>(ui_dst, ui_src, ui_w, u1, i2, NNZ_UI);

  // ---- aggregates
  esr_weighted3<<<gU, TB, 0, stream>>>(u0, u1, u2, nw, uw, UD);   // user_weighted
  esr_scale_by_nw<<<gU, TB, 0, stream>>>(u2, nw, 0, sw, UD);      // sw = nw0 * s0

  // ---- social layer 0 (curIn = u2 -> cur1)
  esr_gemm_wmma<<<gGemm, TB, 0, stream>>>(u2, u2, Wsoc, lin, U, 64);
  esr_copy<<<gU, TB, 0, stream>>>(u2, nbuf, UD);
  esr_spmm<<<gEs, TB, 0, stream>>>(s_src, s_dst, s_w, lin, nbuf, NNZ_S);
  esr_gemm_wmma<<<gGemm, TB, 0, stream>>>(nbuf, nbuf, ho_W, lin, U, 64);
  esr_social_epilogue<<<gRow, TB, 0, stream>>>(lin, ho_b, ho_g, ho_beta,
                                               nbuf, u2, cur1, sw, nw, 1, U);

  // ---- social layer 1 (curIn = cur1 -> cur2)
  esr_gemm_wmma<<<gGemm, TB, 0, stream>>>(cur1, cur1, Wsoc + (size_t)D * D, lin, U, 64);
  esr_copy<<<gU, TB, 0, stream>>>(cur1, nbuf, UD);
  esr_spmm<<<gEs, TB, 0, stream>>>(s_src, s_dst, s_w, lin, nbuf, NNZ_S);
  esr_gemm_wmma<<<gGemm, TB, 0, stream>>>(nbuf, nbuf, ho_W, lin, U, 64);
  esr_social_epilogue<<<gRow, TB, 0, stream>>>(lin, ho_b, ho_g, ho_beta,
                                               nbuf, cur1, cur2, sw, nw, 2, U);

  // ---- metapath fusion: [uw | sw] @ mp_W  (K = 128)
  esr_gemm_wmma<<<gGemm, TB, 0, stream>>>(uw, sw, mp_W, lin, U, 128);
  esr_mp_epilogue<<<gRow, TB, 0, stream>>>(lin, mp_b, mp_g, mp_beta, nbuf, U);

  // ---- scoring
  const int gS = (B * 32 + TB - 1) / TB;
  esr_score<<<gS, TB, 0, stream>>>(users, pos_items, nbuf, i0, i1, i2, nw,
                                   (float*)d_out, B);
}